// S4DKernel_14422500179971
// MI455X (gfx1250) — compile-verified
//
#include <hip/hip_runtime.h>
#include <math.h>

typedef float v2f __attribute__((ext_vector_type(2)));
typedef float v8f __attribute__((ext_vector_type(8)));

#define D_MODEL 1024
#define D_STATE 32
#define L_LEN   2048

// One block per channel h. 256 threads = 8 waves (wave32).
// Wave c computes K[h, m*128 + (16c..16c+15)] for all m=0..15 via
//   Re(U)@Re(V) + (-Im(U))@Im(V)   using V_WMMA_F32_16X16X4_F32, K=32 in 8 chunks.
__global__ __launch_bounds__(256) void s4d_wmma_kernel(
    const float* __restrict__ log_dt,
    const float* __restrict__ log_A_real,
    const float* __restrict__ A_imag,
    const float* __restrict__ B_re, const float* __restrict__ B_im,
    const float* __restrict__ C_re, const float* __restrict__ C_im,
    float* __restrict__ out)
{
    __shared__ float s_are[D_STATE], s_aim[D_STATE];  // dtA (discrete)
    __shared__ float s_wr[D_STATE],  s_wi[D_STATE];   // W = C * B_bar

    const int h    = blockIdx.x;
    const int tid  = threadIdx.x;
    const int lane = tid & 31;         // wave32
    const int wave = tid >> 5;         // column-tile index c = 0..7
    const int col  = lane & 15;
    const int hi   = lane >> 4;        // lane half selects K pair / M high rows

    // ---- per-channel prep: wave 0 (lanes == tid here), no divergence ----
    if (tid < D_STATE) {
        const int n = tid;
        const float dt  = __expf(log_dt[h]);
        const float Are = -__expf(log_A_real[h * D_STATE + n]); // continuous A
        const float Aim = A_imag[h * D_STATE + n];
        const float are = Are * dt;                              // dtA
        const float aim = Aim * dt;
        // A_bar = exp(dtA)
        float s, c;
        const float er = __expf(are);
        __sincosf(aim, &s, &c);
        const float nr = er * c - 1.0f;      // A_bar - 1 (real)
        const float ni = er * s;             //            (imag)
        // (A_bar - 1)/A  via conj(A)/|A|^2
        const float inv = 1.0f / (Are * Are + Aim * Aim);
        const float Bbr = (nr * Are + ni * Aim) * inv;
        const float Bbi = (ni * Are - nr * Aim) * inv;
        // * B
        const float br = B_re[h * D_STATE + n], bi = B_im[h * D_STATE + n];
        const float tr = Bbr * br - Bbi * bi;
        const float ti = Bbr * bi + Bbi * br;
        // W = C * B_bar
        const float cr = C_re[h * D_STATE + n], ci = C_im[h * D_STATE + n];
        s_wr[n]  = cr * tr - ci * ti;
        s_wi[n]  = cr * ti + ci * tr;
        s_are[n] = are;
        s_aim[n] = aim;
    }
    __syncthreads();

    const float tA = 128.0f * (float)col;          // A-frag: M = col (= l_hi), t = 128*m
    const float tB = (float)(wave * 16 + col);     // B-frag: column = l_lo

    v8f acc = {};
    #pragma unroll
    for (int chunk = 0; chunk < 8; ++chunk) {
        v2f Ar, Ain, Br, Bi;
        #pragma unroll
        for (int j = 0; j < 2; ++j) {
            // ISA 16x4 f32 A layout: VGPR j holds K = 2*hi + j (per K-chunk base)
            const int n = chunk * 4 + hi * 2 + j;
            const float are = s_are[n], aim = s_aim[n];
            const float wr  = s_wr[n],  wi  = s_wi[n];
            // U[m,n] = W[n] * exp(dtA[n] * 128m)
            float sa, ca;
            const float ea = __expf(are * tA);
            __sincosf(aim * tA, &sa, &ca);
            const float pr = ea * ca, pi = ea * sa;
            Ar[j]  = wr * pr - wi * pi;            // Re(U)
            Ain[j] = -(wr * pi + wi * pr);         // -Im(U)
            // V[n,j] = exp(dtA[n] * l_lo)
            float sb, cb;
            const float eb = __expf(are * tB);
            __sincosf(aim * tB, &sb, &cb);
            Br[j] = eb * cb;                       // Re(V)
            Bi[j] = eb * sb;                       // Im(V)
        }
        // D = A*B + C, f32 in / f32 accumulate (16x16x4)
        acc = __builtin_amdgcn_wmma_f32_16x16x4_f32(
                  false, Ar,  false, Br, (short)0, acc, false, false);
        acc = __builtin_amdgcn_wmma_f32_16x16x4_f32(
                  false, Ain, false, Bi, (short)0, acc, false, false);
    }

    // D layout: VGPR r -> M = r + 8*hi, N = col. l = M*128 + wave*16 + col.
    float* op = out + (size_t)h * L_LEN + wave * 16 + col;
    #pragma unroll
    for (int r = 0; r < 8; ++r) {
        op[(size_t)(r + hi * 8) * 128] = acc[r];
    }
}

extern "C" void kernel_launch(void* const* d_in, const int* in_sizes, int n_in,
                              void* d_out, int out_size, void* d_ws, size_t ws_size,
                              hipStream_t stream) {
    const float* log_dt     = (const float*)d_in[0];
    const float* log_A_real = (const float*)d_in[1];
    const float* A_imag     = (const float*)d_in[2];
    const float* B_re       = (const float*)d_in[3];
    const float* B_im       = (const float*)d_in[4];
    const float* C_re       = (const float*)d_in[5];
    const float* C_im       = (const float*)d_in[6];
    float* out = (float*)d_out;

    s4d_wmma_kernel<<<dim3(D_MODEL), dim3(256), 0, stream>>>(
        log_dt, log_A_real, A_imag, B_re, B_im, C_re, C_im, out);
}